// PLEmbedding_58961311039690
// MI455X (gfx1250) — compile-verified
//
#include <hip/hip_runtime.h>
#include <stdint.h>

// Address-space helpers for gfx1250 async global->LDS builtins.
#define GAS __attribute__((address_space(1)))
#define LAS __attribute__((address_space(3)))

#if __has_builtin(__builtin_amdgcn_global_load_async_to_lds_b32) && \
    __has_builtin(__builtin_amdgcn_s_wait_asynccnt)
#define PLE_ASYNC_LDS 1
#else
#define PLE_ASYNC_LDS 0
#endif

static constexpr int kD = 64;          // embedding dim (bins count)
static constexpr int kGroups = kD / 4; // float4 groups per x element = 16

// out = (x < hi) ? ((x >= lo) ? (x-lo)/(hi-lo) : 0) : 1
// Division replaced by v_rcp_f32 + mul (TRANS op, co-executes with VALU).
__device__ __forceinline__ float ple_piece(float xv, float lo, float hi) {
  float frac = (xv - lo) * __builtin_amdgcn_rcpf(hi - lo);
  return (xv < hi) ? ((xv >= lo) ? frac : 0.0f) : 1.0f;
}

// Layout: gid = xi*16 + g. Each thread produces out[xi, g*4 .. g*4+3] as one
// float4 -> consecutive lanes write consecutive 16B -> global_store_b128,
// fully coalesced (4 KB contiguous per 256-thread block).
__global__ __launch_bounds__(256) void ple_kernel(const float* __restrict__ x,
                                                  const float* __restrict__ bins,
                                                  float* __restrict__ out,
                                                  int n_x) {
  __shared__ float sbins[kD + 1];
  const int t = threadIdx.x;

#if PLE_ASYNC_LDS
  // Stage the 65-entry bin table via the CDNA5 async copy path (ASYNCcnt).
  // Builtin signature (per hipcc diagnostic): (global int*, local int*, imm, imm)
  if (t <= kD) {
    __builtin_amdgcn_global_load_async_to_lds_b32(
        (GAS int*)(bins + t), (LAS int*)(sbins + t), /*offset=*/0, /*cpol=*/0);
  }
  __builtin_amdgcn_s_wait_asynccnt(0);
#else
  if (t <= kD) sbins[t] = bins[t];
#endif
  __syncthreads();

  const int gid = blockIdx.x * 256 + t;
  const int xi = gid >> 4;           // 16 lanes per x element
  if (xi >= n_x) return;
  const int g = gid & (kGroups - 1); // which float4 of the 64-wide row

  const float xv = x[xi];            // 16-way redundant load, one 64B line/block

  const int j0 = g * 4;
  // LDS reads: lanes 16..31 mirror lanes 0..15 (same addresses -> broadcast).
  const float b0 = sbins[j0 + 0];
  const float b1 = sbins[j0 + 1];
  const float b2 = sbins[j0 + 2];
  const float b3 = sbins[j0 + 3];
  const float b4 = sbins[j0 + 4];

  float4 o;
  o.x = ple_piece(xv, b0, b1);
  o.y = ple_piece(xv, b1, b2);
  o.z = ple_piece(xv, b2, b3);
  o.w = ple_piece(xv, b3, b4);

  float4* op = reinterpret_cast<float4*>(out + ((size_t)xi * kD + j0));
  *op = o;
}

extern "C" void kernel_launch(void* const* d_in, const int* in_sizes, int n_in,
                              void* d_out, int out_size, void* d_ws, size_t ws_size,
                              hipStream_t stream) {
  const float* x = (const float*)d_in[0];     // (B, L) f32
  const float* bins = (const float*)d_in[1];  // (D+1,) f32
  float* out = (float*)d_out;                 // (B, L, D) f32
  (void)n_in; (void)out_size; (void)d_ws; (void)ws_size;

  const int n_x = in_sizes[0];                // B*L
  const long long total_threads = (long long)n_x * kGroups;
  const int block = 256;                       // 8 wave32 waves per block
  const int grid = (int)((total_threads + block - 1) / block);

  ple_kernel<<<grid, block, 0, stream>>>(x, bins, out, n_x);
}